// VQVAELayer_10471130267722
// MI455X (gfx1250) — compile-verified
//
#include <hip/hip_runtime.h>

typedef __attribute__((ext_vector_type(2))) float v2f;
typedef __attribute__((ext_vector_type(8))) float v8f;

#define NROWS 32768
#define DIM   64
#define KCB   8192
#define KTILE 128
#define LSTR  68   // LDS row stride in floats (64 + 4 pad -> conflict-free A reads)

// ---------------- |w_k|^2 precompute ----------------
__global__ __launch_bounds__(256) void wsq_kernel(const float* __restrict__ w,
                                                  float* __restrict__ wsq) {
  int k = blockIdx.x * 256 + threadIdx.x;
  float s = 0.0f;
#pragma unroll
  for (int d = 0; d < DIM; ++d) {
    float v = w[d * KCB + k];
    s += v * v;
  }
  wsq[k] = s;
}

// ---------------- main VQ kernel: WMMA fp32 GEMM + fused argmin + gather ----------------
__global__ __launch_bounds__(256) void vq_kernel(const float* __restrict__ x,
                                                 const float* __restrict__ w,
                                                 const float* __restrict__ wsq,
                                                 float* __restrict__ out) {
  __shared__ float lds_w[KTILE * LSTR];   // staged w tile, transposed: [kc][d]
  __shared__ float lds_q[KTILE];          // staged |w|^2 tile

  const int tid  = threadIdx.x;
  const int lane = tid & 31;
  const int wave = tid >> 5;          // 0..7 -> 8 waves per block
  const int half = lane >> 4;         // 0 (lanes 0-15) / 1 (lanes 16-31)
  const int l16  = lane & 15;
  const int c0   = half * 2;          // A/B K-offset for upper half-wave

  const int nbase = blockIdx.x * 128 + wave * 16;  // 16 x-rows per wave
  const int n     = nbase + l16;

  // Preload B operand (this lane's x row, the 32 dims it needs) into registers.
  v2f bx[16];
#pragma unroll
  for (int t = 0; t < 16; ++t) {
    const float* p = x + n * DIM + 4 * t + c0;
    v2f b; b.x = p[0]; b.y = p[1];
    bx[t] = b;
  }

  float best    = 3.402823466e38f;
  int   bestIdx = 0;

  for (int k0 = 0; k0 < KCB; k0 += KTILE) {
    __syncthreads();
    // Stage w[0:64, k0:k0+128] transposed into LDS (global reads coalesced over k).
    {
      const int kc = tid & 127;
      for (int d = (tid >> 7); d < DIM; d += 2)
        lds_w[kc * LSTR + d] = w[d * KCB + k0 + kc];
      if (tid < KTILE) lds_q[tid] = wsq[k0 + tid];
    }
    __syncthreads();

#pragma unroll
    for (int sub = 0; sub < KTILE / 16; ++sub) {
      const float* acol = &lds_w[(sub * 16 + l16) * LSTR + c0];
      v8f acc = {0.f, 0.f, 0.f, 0.f, 0.f, 0.f, 0.f, 0.f};
#pragma unroll
      for (int t = 0; t < 16; ++t) {
        v2f a; a.x = acol[4 * t]; a.y = acol[4 * t + 1];
        // D(16x16 f32) += A(16x4 f32: codes x dims) * B(4x16 f32: dims x rows)
        acc = __builtin_amdgcn_wmma_f32_16x16x4_f32(
            /*neg_a=*/false, a, /*neg_b=*/false, bx[t],
            /*c_mod=*/(short)0, acc, /*reuse_a=*/false, /*reuse_b=*/false);
      }
      // Scores for codes k = k0 + sub*16 + half*8 + j  (rows = this lane's N)
      const int kb = k0 + sub * 16 + half * 8;
#pragma unroll
      for (int j = 0; j < 8; ++j) {
        float s = lds_q[sub * 16 + half * 8 + j] - 2.0f * acc[j];
        int   kk = kb + j;
        if (s < best) { best = s; bestIdx = kk; }
      }
    }
  }

  // Combine the two half-wave candidates for each x-row (lane n <-> lane n+16).
  {
    float ob = __shfl_xor(best, 16, 32);
    int   oi = __shfl_xor(bestIdx, 16, 32);
    if (ob < best || (ob == best && oi < bestIdx)) { best = ob; bestIdx = oi; }
  }

  // Gather: out[row] = w[:, bestIdx(row)], coalesced writes (whole wave per row).
#pragma unroll
  for (int r = 0; r < 16; ++r) {
    int idx = __shfl(bestIdx, r, 32);
    int row = nbase + r;
    out[row * DIM + lane]      = w[lane * KCB + idx];
    out[row * DIM + lane + 32] = w[(lane + 32) * KCB + idx];
  }
}

extern "C" void kernel_launch(void* const* d_in, const int* in_sizes, int n_in,
                              void* d_out, int out_size, void* d_ws, size_t ws_size,
                              hipStream_t stream) {
  const float* x = (const float*)d_in[0];   // [N, D]
  const float* w = (const float*)d_in[1];   // [D, K]
  float* out = (float*)d_out;               // [N, D]
  float* wsq = (float*)d_ws;                // K floats scratch

  wsq_kernel<<<KCB / 256, 256, 0, stream>>>(w, wsq);
  vq_kernel<<<NROWS / 128, 256, 0, stream>>>(x, w, wsq, out);
}